// IRSE_22591527976998
// MI455X (gfx1250) — compile-verified
//
#include <hip/hip_runtime.h>
#include <math.h>

// ---------------------------------------------------------------------------
// CDNA5 / gfx1250, wave32. fp32 WMMA: V_WMMA_F32_16X16X4_F32
//   A: 16x4 f32  -> 2 VGPRs (v2f)   lane: M=l&15, v0=K(4k+2h), v1=K+1, h=l>>4
//   B: 4x16 f32  -> 2 VGPRs (v2f)   lane: N=l&15, v0=K(4k+2h), v1=K+1
//   C/D: 16x16   -> 8 VGPRs (v8f)   vgpr r: M=r+8h, N=l&15
// ---------------------------------------------------------------------------
typedef __attribute__((ext_vector_type(2))) float v2f;
typedef __attribute__((ext_vector_type(8))) float v8f;

__device__ __forceinline__ v8f wmma_f32(v2f a, v2f b, v8f c) {
    // 8 args: (neg_a, A, neg_b, B, c_mod, C, reuse_a, reuse_b)
    return __builtin_amdgcn_wmma_f32_16x16x4_f32(
        false, a, false, b, (short)0, c, false, false);
}

// D[k][n] = 0.5*cos(pi*(2n+1)k/16), row 0 scaled by 1/sqrt(2).
// Compile-time table: avoids libm cosf (Payne-Hanek path) in device code.
__device__ __constant__ float c_D[64] = {
    0.3535533906f,  0.3535533906f,  0.3535533906f,  0.3535533906f,
    0.3535533906f,  0.3535533906f,  0.3535533906f,  0.3535533906f,

    0.4903926402f,  0.4157348062f,  0.2777851165f,  0.0975451610f,
   -0.0975451610f, -0.2777851165f, -0.4157348062f, -0.4903926402f,

    0.4619397663f,  0.1913417162f, -0.1913417162f, -0.4619397663f,
   -0.4619397663f, -0.1913417162f,  0.1913417162f,  0.4619397663f,

    0.4157348062f, -0.0975451610f, -0.4903926402f, -0.2777851165f,
    0.2777851165f,  0.4903926402f,  0.0975451610f, -0.4157348062f,

    0.3535533906f, -0.3535533906f, -0.3535533906f,  0.3535533906f,
    0.3535533906f, -0.3535533906f, -0.3535533906f,  0.3535533906f,

    0.2777851165f, -0.4903926402f,  0.0975451610f,  0.4157348062f,
   -0.4157348062f, -0.0975451610f,  0.4903926402f, -0.2777851165f,

    0.1913417162f, -0.4619397663f,  0.4619397663f, -0.1913417162f,
   -0.1913417162f,  0.4619397663f, -0.4619397663f,  0.1913417162f,

    0.0975451610f, -0.2777851165f,  0.4157348062f, -0.4903926402f,
    0.4903926402f, -0.4157348062f,  0.2777851165f, -0.0975451610f
};

// D2 = diag(D, D) (16x16 block-diagonal): one cached load + select
__device__ __forceinline__ float d2(int r, int c) {
    const float v = c_D[((r & 7) << 3) | (c & 7)];
    return ((r >> 3) == (c >> 3)) ? v : 0.0f;
}

// ---------------------------------------------------------------------------
// Forward DCT: per 16x16 image tile (2x2 blocks), out = D2 * X * D2^T,
// scattered into feat[f][bj][bi]  (f = i*8+j)
// grid 64 x 128 threads (4 waves/block), one tile per wave, 256 tiles
// ---------------------------------------------------------------------------
__global__ void __launch_bounds__(128)
irse_dct_kernel(const float* __restrict__ x, float* __restrict__ feat) {
    __shared__ float lds[4 * 256];
    const int lane = threadIdx.x & 31;
    const int wave = threadIdx.x >> 5;
    const int tile = blockIdx.x * 4 + wave;
    const int ti = tile >> 4, tj = tile & 15;   // row-tile, col-tile
    const int h = lane >> 4;                    // lane half
    const int N = lane & 15;                    // column / A-row index
    const int row0 = ti * 16, col0 = tj * 16;

    // pass 1: tmp = D2 * X
    v8f acc = {};
#pragma unroll
    for (int k = 0; k < 4; ++k) {
        const int ka = 4 * k + 2 * h;
        v2f a, b;
        a.x = d2(N, ka);
        a.y = d2(N, ka + 1);
        // img[r][c] = x[(r*256+c)*3] (channel 0)
        b.x = x[(((row0 + ka) * 256) + col0 + N) * 3];
        b.y = x[(((row0 + ka + 1) * 256) + col0 + N) * 3];
        acc = wmma_f32(a, b, acc);
    }

    // transpose tmp via LDS: C/D layout -> A layout
    float* wlds = lds + wave * 256;
#pragma unroll
    for (int r = 0; r < 8; ++r) wlds[(r + 8 * h) * 16 + N] = acc[r];
    __syncthreads();

    // pass 2: out = tmp * D2^T   (B[K][N] = D2[N][K])
    v8f acc2 = {};
#pragma unroll
    for (int k = 0; k < 4; ++k) {
        const int ka = 4 * k + 2 * h;
        v2f a, b;
        a.x = wlds[N * 16 + ka];
        a.y = wlds[N * 16 + ka + 1];
        b.x = d2(N, ka);
        b.y = d2(N, ka + 1);
        acc2 = wmma_f32(a, b, acc2);
    }

    // scatter: element (M,N) -> feat[f=(M&7)*8+(N&7)][bj=2tj+(N>>3)][bi=2ti+(M>>3)]
#pragma unroll
    for (int r = 0; r < 8; ++r) {
        const int M = r + 8 * h;
        const int i = M & 7, bi = 2 * ti + (M >> 3);
        const int j = N & 7, bj = 2 * tj + (N >> 3);
        feat[(i * 8 + j) * 1024 + bj * 32 + bi] = acc2[r];
    }
}

// ---------------------------------------------------------------------------
// Channel squeeze-excite MLP: sq = mean(feat) over 1024 blocks per channel,
// h = relu(sq @ fc1_w^T + fc1_b), scale_c = sigmoid(h @ fc2_w^T + fc2_b)
// one block, 256 threads
// ---------------------------------------------------------------------------
__global__ void __launch_bounds__(256)
irse_se_mlp_kernel(const float* __restrict__ feat,
                   const float* __restrict__ fc1_w, const float* __restrict__ fc1_b,
                   const float* __restrict__ fc2_w, const float* __restrict__ fc2_b,
                   float* __restrict__ scale_c) {
    __shared__ float part[256];
    __shared__ float sq[64];
    __shared__ float hbuf[32];
    const int t = threadIdx.x;
    const int f = t >> 2, q = t & 3;
    float s = 0.0f;
    const float* p = feat + f * 1024 + q * 256;
    for (int i = 0; i < 256; ++i) s += p[i];
    part[t] = s;
    __syncthreads();
    if (q == 0)
        sq[f] = (part[t] + part[t + 1] + part[t + 2] + part[t + 3]) * (1.0f / 1024.0f);
    __syncthreads();
    if (t < 32) {
        float acc = fc1_b[t];
        for (int c = 0; c < 64; ++c) acc += sq[c] * fc1_w[t * 64 + c];
        hbuf[t] = fmaxf(acc, 0.0f);
    }
    __syncthreads();
    if (t < 64) {
        float acc = fc2_b[t];
        for (int r = 0; r < 32; ++r) acc += hbuf[r] * fc2_w[t * 32 + r];
        scale_c[t] = 1.0f / (1.0f + expf(-acc));
    }
}

// ---------------------------------------------------------------------------
// Spatial gate: spatial[h,w] = sigmoid(sum_f sse_w[f]*feat[f,h,w] + sse_b)
// 4 blocks x 256 threads = 1024 positions
// ---------------------------------------------------------------------------
__global__ void __launch_bounds__(256)
irse_spatial_kernel(const float* __restrict__ feat, const float* __restrict__ sse_w,
                    const float* __restrict__ sse_b, float* __restrict__ spatial) {
    const int idx = blockIdx.x * 256 + threadIdx.x;   // bj*32 + bi
    float acc = sse_b[0];
    for (int f = 0; f < 64; ++f) acc += sse_w[f] * feat[f * 1024 + idx];
    spatial[idx] = 1.0f / (1.0f + expf(-acc));
}

// ---------------------------------------------------------------------------
// Inverse DCT with fused gating: Y = max(feat*scale_c, feat*spatial),
// out_tile = D2^T * Y * D2, written to 256x256 scratch plane
// ---------------------------------------------------------------------------
__device__ __forceinline__ float irse_yval(const float* __restrict__ feat,
                                           const float* __restrict__ sc,
                                           const float* __restrict__ sp,
                                           int K, int N, int ti, int tj) {
    const int f = (K & 7) * 8 + (N & 7);
    const int bj = 2 * tj + (N >> 3);
    const int bi = 2 * ti + (K >> 3);
    const float v = feat[f * 1024 + bj * 32 + bi];
    return fmaxf(v * sc[f], v * sp[bj * 32 + bi]);
}

__global__ void __launch_bounds__(128)
irse_idct_kernel(const float* __restrict__ feat, const float* __restrict__ scale_c,
                 const float* __restrict__ spatial, float* __restrict__ out2d) {
    __shared__ float lds[4 * 256];
    const int lane = threadIdx.x & 31;
    const int wave = threadIdx.x >> 5;
    const int tile = blockIdx.x * 4 + wave;
    const int ti = tile >> 4, tj = tile & 15;
    const int h = lane >> 4;
    const int N = lane & 15;

    // pass 1: tmp = D2^T * Y     (A[M][K] = D2[K][M])
    v8f acc = {};
#pragma unroll
    for (int k = 0; k < 4; ++k) {
        const int ka = 4 * k + 2 * h;
        v2f a, b;
        a.x = d2(ka, N);
        a.y = d2(ka + 1, N);
        b.x = irse_yval(feat, scale_c, spatial, ka, N, ti, tj);
        b.y = irse_yval(feat, scale_c, spatial, ka + 1, N, ti, tj);
        acc = wmma_f32(a, b, acc);
    }

    float* wlds = lds + wave * 256;
#pragma unroll
    for (int r = 0; r < 8; ++r) wlds[(r + 8 * h) * 16 + N] = acc[r];
    __syncthreads();

    // pass 2: out = tmp * D2     (B[K][N] = D2[K][N])
    v8f acc2 = {};
#pragma unroll
    for (int k = 0; k < 4; ++k) {
        const int ka = 4 * k + 2 * h;
        v2f a, b;
        a.x = wlds[N * 16 + ka];
        a.y = wlds[N * 16 + ka + 1];
        b.x = d2(ka, N);
        b.y = d2(ka + 1, N);
        acc2 = wmma_f32(a, b, acc2);
    }

    const int row0 = ti * 16, col0 = tj * 16;
#pragma unroll
    for (int r = 0; r < 8; ++r) {
        const int M = r + 8 * h;
        out2d[(row0 + M) * 256 + col0 + N] = acc2[r];
    }
}

// ---------------------------------------------------------------------------
// Broadcast: (256,256) plane -> (64,3,256,256). 48 MB of b128 stores; this is
// the only bandwidth-relevant kernel (~2.1 us at 23.3 TB/s). Source stays in L2.
// ---------------------------------------------------------------------------
__global__ void __launch_bounds__(256)
irse_bcast_kernel(const float4* __restrict__ out2d4, float4* __restrict__ out4) {
    const unsigned idx = blockIdx.x * 256u + threadIdx.x;  // 0 .. 3145727
    out4[idx] = out2d4[idx & 16383u];                      // 65536/4 per plane
}

// ---------------------------------------------------------------------------
extern "C" void kernel_launch(void* const* d_in, const int* in_sizes, int n_in,
                              void* d_out, int out_size, void* d_ws, size_t ws_size,
                              hipStream_t stream) {
    const float* x     = (const float*)d_in[0];
    const float* fc1_w = (const float*)d_in[1];
    const float* fc1_b = (const float*)d_in[2];
    const float* fc2_w = (const float*)d_in[3];
    const float* fc2_b = (const float*)d_in[4];
    const float* sse_w = (const float*)d_in[5];
    const float* sse_b = (const float*)d_in[6];
    // d_in[7] = bs (fixed 64; output shape baked into out_size)

    float* ws      = (float*)d_ws;
    float* feat    = ws;                 // 64*32*32        = 65536 floats (256 KB)
    float* out2d   = ws + 65536;         // 256*256         = 65536 floats (256 KB)
    float* scale_c = ws + 131072;        // 64 floats
    float* spatial = ws + 131072 + 64;   // 1024 floats

    irse_dct_kernel<<<64, 128, 0, stream>>>(x, feat);
    irse_se_mlp_kernel<<<1, 256, 0, stream>>>(feat, fc1_w, fc1_b, fc2_w, fc2_b, scale_c);
    irse_spatial_kernel<<<4, 256, 0, stream>>>(feat, sse_w, sse_b, spatial);
    irse_idct_kernel<<<64, 128, 0, stream>>>(feat, scale_c, spatial, out2d);
    // 64*3*256*256 / 4 = 3,145,728 float4 stores
    irse_bcast_kernel<<<12288, 256, 0, stream>>>((const float4*)out2d, (float4*)d_out);
}